// MultiClusterModel2_80822694576388
// MI455X (gfx1250) — compile-verified
//
#include <hip/hip_runtime.h>
#include <hip/hip_bf16.h>
#include <math.h>

// ---------------- model constants ----------------
#define BB   128
#define INL  1984
#define OUTL 400
#define C0   20
#define CC   100
#define K0   51
#define KD   7
#define WH   1500
#define KEYL 16
#define ZD   2
#define NB   7
#define NQ   (NB*CC)      // 700
#define L0   992          // after stride-2 layer0
#define LIN_IN 248
#define QPW  4            // queries per wave in warehouse kernel

typedef float v2f __attribute__((ext_vector_type(2)));
typedef float v8f __attribute__((ext_vector_type(8)));

__device__ __forceinline__ float lrelu(float v) { return fmaxf(v, 0.2f * v); }

// ---------------- layer0 conv a: (B,5,1984) -> (B,20,992), stride2, pad25, +bias, lrelu ----
__global__ void k_conv0a(const float* __restrict__ x, const float* __restrict__ w,
                         const float* __restrict__ bias, float* __restrict__ y, int total)
{
    int t = blockIdx.x * blockDim.x + threadIdx.x;
    if (t >= total) return;
    int l = t % L0;
    int co = (t / L0) % C0;
    int b = t / (L0 * C0);
    float v = bias[co];
    const float* xb = x + (size_t)b * 5 * INL;
    const float* wc = w + co * 5 * K0;
    for (int ci = 0; ci < 5; ci++) {
        const float* xc = xb + ci * INL;
        const float* wk = wc + ci * K0;
        #pragma unroll 17
        for (int kk = 0; kk < K0; kk++) {
            int li = 2 * l + kk - 25;
            int liC = min(max(li, 0), INL - 1);
            float msk = (li >= 0 && li < INL) ? 1.f : 0.f;
            v += wk[kk] * xc[liC] * msk;
        }
    }
    y[t] = lrelu(v);
}

// ---------------- layer0 conv b as WMMA im2col GEMM ------------------------------
// out(co, b*l) = sum_{t=0..50} sum_{ci=0..19} W[co,ci,t] * X[b,ci,l+t-25]
// tap loop outside, K=20 channel GEMM inside; pad handled by clamp+mask (no branch)
__global__ __launch_bounds__(32)
void k_conv0b_wmma(const float* __restrict__ W,   // (20,20,51)
                   const float* __restrict__ X,   // (B,20,992)
                   const float* __restrict__ g, const float* __restrict__ bta,
                   float* __restrict__ Y)         // (B,20,992)
{
    const int ntilesL = L0 / 16;                  // 62
    int b  = blockIdx.x / ntilesL;
    int lt = blockIdx.x % ntilesL;
    int mBase = blockIdx.y * 16;                  // 0 or 16
    int lane = threadIdx.x;
    int h = lane >> 4, r = lane & 15;
    int col = lt * 16 + r;                        // < 992 always
    int mRow = mBase + r;
    int mRowC = mRow < C0 ? mRow : C0 - 1;        // clamp padded rows (masked at store)
    const float* Xb = X + (size_t)b * C0 * L0;
    const float* Wr = W + mRowC * (C0 * K0);
    v8f c = {0.f,0.f,0.f,0.f,0.f,0.f,0.f,0.f};
    for (int t = 0; t < K0; t++) {
        int li = col + t - 25;
        bool in = (li >= 0) & (li < L0);
        int liC = in ? li : 0;
        float msk = in ? 1.f : 0.f;
        const float* xt = Xb + liC;
        const float* wt = Wr + t;
        #pragma unroll
        for (int k0 = 0; k0 < C0; k0 += 4) {
            int ka = k0 + 2 * h;                  // A/B: K = 2*half + vgpr
            v2f a, bb;
            a.x  = wt[ka * K0];
            a.y  = wt[(ka + 1) * K0];
            bb.x = xt[ka * L0] * msk;
            bb.y = xt[(ka + 1) * L0] * msk;
            c = __builtin_amdgcn_wmma_f32_16x16x4_f32(false, a, false, bb, (short)0, c,
                                                      false, false);
        }
    }
    #pragma unroll
    for (int rr = 0; rr < 8; rr++) {
        int m = mBase + rr + 8 * h;               // C/D: M = rr + 8*half
        if (m < C0) {
            float v = g[m] * c[rr] + bta[m];
            Y[((size_t)b * C0 + m) * L0 + col] = lrelu(v);
        }
    }
}

// ---------------- query decoder: z=mu[cid] -> h1 -> h2 (tanh MLP) ----------------
__global__ void k_qd(const int* __restrict__ cid, const float* __restrict__ vae_mu,
                     const float* __restrict__ w1, const float* __restrict__ b1,
                     const float* __restrict__ w2, const float* __restrict__ b2,
                     float* __restrict__ h2out)
{
    int b = blockIdx.x;
    int t = threadIdx.x;
    __shared__ float h1[30];
    int c = cid[b];
    float z0 = vae_mu[c * ZD + 0];
    float z1 = vae_mu[c * ZD + 1];
    if (t < 30) h1[t] = tanhf(w1[t * 2] * z0 + w1[t * 2 + 1] * z1 + b1[t]);
    __syncthreads();
    if (t < 30) {
        float v = b2[t];
        #pragma unroll
        for (int k = 0; k < 30; k++) v += w2[t * 30 + k] * h1[k];
        h2out[b * 30 + t] = tanhf(v);
    }
}

// queries[b, i] = h2[b,:] . qd_w3[i,:] + qd_b3[i]   (i in [0, 11200))
__global__ void k_queries(const float* __restrict__ h2, const float* __restrict__ w3,
                          const float* __restrict__ b3, float* __restrict__ q, int total)
{
    int t = blockIdx.x * blockDim.x + threadIdx.x;
    if (t >= total) return;
    int i = t % (NQ * KEYL);
    int b = t / (NQ * KEYL);
    float v = b3[i];
    const float* h = h2 + b * 30;
    const float* w = w3 + (size_t)i * 30;
    #pragma unroll
    for (int k = 0; k < 30; k++) v += h[k] * w[k];
    q[t] = v;
}

// z_mu / z_logvar gather to output tail
__global__ void k_z(const int* __restrict__ cid, const float* __restrict__ mu,
                    const float* __restrict__ lv, float* __restrict__ omu, float* __restrict__ olv)
{
    int t = blockIdx.x * blockDim.x + threadIdx.x;
    if (t >= BB * ZD) return;
    int b = t / ZD, d = t % ZD;
    int c = cid[b];
    omu[t] = mu[c * ZD + d];
    olv[t] = lv[c * ZD + d];
}

// ---------------- warehouse attention, fused online softmax ----------------------
// one wave handles QPW consecutive (b,q) rows; key/value rows loaded once per wave
__global__ void k_warehouse(const float* __restrict__ queries, const float* __restrict__ keys,
                            const float* __restrict__ whw, const float* __restrict__ whb,
                            float* __restrict__ dynW, float* __restrict__ dynb)
{
    int wv = blockIdx.x * (blockDim.x >> 5) + (threadIdx.x >> 5);
    int row0 = wv * QPW;
    int lane = threadIdx.x & 31;
    float qv[QPW][KEYL];
    #pragma unroll
    for (int qq = 0; qq < QPW; qq++)
        #pragma unroll
        for (int j = 0; j < KEYL; j++) qv[qq][j] = queries[(size_t)(row0 + qq) * KEYL + j];
    float m[QPW], s[QPW], acc[QPW][8];
    #pragma unroll
    for (int qq = 0; qq < QPW; qq++) {
        m[qq] = -INFINITY; s[qq] = 0.f;
        #pragma unroll
        for (int t = 0; t < 8; t++) acc[qq][t] = 0.f;
    }
    for (int k = lane; k < WH; k += 32) {
        float kr[KEYL];
        #pragma unroll
        for (int j = 0; j < KEYL; j++) kr[j] = keys[k * KEYL + j];
        float w7[7];
        #pragma unroll
        for (int t = 0; t < 7; t++) w7[t] = whw[k * KD + t];
        float wb = whb[k];
        #pragma unroll
        for (int qq = 0; qq < QPW; qq++) {
            float d = 0.f;
            #pragma unroll
            for (int j = 0; j < KEYL; j++) d += qv[qq][j] * kr[j];
            d *= 0.25f;
            float nm = fmaxf(m[qq], d);
            float sc = __expf(m[qq] - nm);
            float e  = __expf(d - nm);
            s[qq] = s[qq] * sc + e;
            #pragma unroll
            for (int t = 0; t < 7; t++) acc[qq][t] = acc[qq][t] * sc + e * w7[t];
            acc[qq][7] = acc[qq][7] * sc + e * wb;
            m[qq] = nm;
        }
    }
    #pragma unroll
    for (int qq = 0; qq < QPW; qq++) {
        float mm = m[qq], ss = s[qq];
        for (int off = 16; off >= 1; off >>= 1) {
            float mo = __shfl_xor(mm, off, 32);
            float so = __shfl_xor(ss, off, 32);
            float nm = fmaxf(mm, mo);
            float s1 = __expf(mm - nm), s2 = __expf(mo - nm);
            ss = ss * s1 + so * s2;
            #pragma unroll
            for (int t = 0; t < 8; t++) {
                float ao = __shfl_xor(acc[qq][t], off, 32);
                acc[qq][t] = acc[qq][t] * s1 + ao * s2;
            }
            mm = nm;
        }
        if (lane == 0) {
            float inv = 1.f / ss;
            #pragma unroll
            for (int t = 0; t < 7; t++) dynW[(size_t)(row0 + qq) * KD + t] = acc[qq][t] * inv;
            dynb[row0 + qq] = acc[qq][7] * inv;
        }
    }
}

// ---------------- WMMA f32 16x16x4 GEMM: Y = epilogue( W(MxK) @ X(K x B*Lout) ) ----
// X layout (B,K,Lin), columns at l*colStride; Y layout (B,M,Lout).
// Branchless: OOB rows/cols are address-clamped (they never contaminate valid
// outputs — WMMA (m,n) depends only on A-row m and B-col n) and masked at store.
template<int RELU, int ADDSHORT, int K>
__global__ __launch_bounds__(32)
void k_gemm(const float* __restrict__ W, int M,
            const float* __restrict__ X, int Lin, int colStride,
            const float* __restrict__ gamma, const float* __restrict__ beta,
            const float* shortBuf,            // (B,M,Lout); may alias Y
            float* Y, int Lout, int ntilesL)
{
    int b  = blockIdx.x / ntilesL;
    int lt = blockIdx.x % ntilesL;
    int mBase = blockIdx.y * 16;
    int lane = threadIdx.x;
    int h = lane >> 4;                 // lane half: selects K pair / M+8
    int r = lane & 15;
    int col = lt * 16 + r;
    bool colOK = col < Lout;
    int xcol = colOK ? col * colStride : 0;      // clamp
    int mRow = mBase + r;
    int mRowC = mRow < M ? mRow : M - 1;         // clamp
    const float* wp = W + mRowC * K + 2 * h;     // wp[k0], wp[k0+1]: 8B-aligned pair
    const float* xp = X + (size_t)b * K * Lin + xcol;
    v8f c = {0.f,0.f,0.f,0.f,0.f,0.f,0.f,0.f};
    #pragma unroll 5
    for (int k0 = 0; k0 < K; k0 += 4) {
        int ka = k0 + 2 * h;                     // A/B: K = 2*half + vgpr
        v2f a = *(const v2f*)(wp + k0);          // one b64 load
        v2f bb;
        bb.x = xp[ka * Lin];
        bb.y = xp[ka * Lin + Lin];
        c = __builtin_amdgcn_wmma_f32_16x16x4_f32(false, a, false, bb, (short)0, c,
                                                  false, false);
    }
    #pragma unroll
    for (int rr = 0; rr < 8; rr++) {
        int mm = mBase + rr + 8 * h;             // C/D: M = rr + 8*half
        if (mm < M && colOK) {
            float v = c[rr];
            v = gamma[mm] * v + beta[mm];
            size_t oidx = ((size_t)b * M + mm) * Lout + col;
            if (ADDSHORT) v += shortBuf[oidx];
            if (RELU) v = lrelu(v);
            Y[oidx] = v;
        }
    }
}

// ---------------- dynamic depthwise K=7 (+dynb) -> affine bn2 -> lrelu -----------
__global__ void k_dw(const float* __restrict__ y1, const float* __restrict__ dynW,
                     const float* __restrict__ dynb, int dynOff,
                     const float* __restrict__ g, const float* __restrict__ bta,
                     float* __restrict__ out, int Lin, int Lout, int stride, int total)
{
    int t = blockIdx.x * blockDim.x + threadIdx.x;
    if (t >= total) return;
    int l = t % Lout;
    int c = (t / Lout) % CC;
    int b = t / (Lout * CC);
    int ch = b * NQ + dynOff + c;
    const float* w = dynW + (size_t)ch * KD;
    float v = dynb[ch];
    const float* xin = y1 + ((size_t)b * CC + c) * Lin;
    int base = l * stride - 3;
    #pragma unroll
    for (int tt = 0; tt < KD; tt++) {
        int li = base + tt;
        int liC = min(max(li, 0), Lin - 1);
        float msk = (li >= 0 && li < Lin) ? 1.f : 0.f;
        v += w[tt] * xin[liC] * msk;
    }
    v = g[c] * v + bta[c];
    out[((size_t)b * CC + c) * Lout + l] = lrelu(v);
}

// ---------------- head: layer4 conv (1,100,3) pad1 + bias -> (B,248) -------------
__global__ void k_head1(const float* __restrict__ x, const float* __restrict__ w,
                        const float* __restrict__ bias, float* __restrict__ y, int total)
{
    int t = blockIdx.x * blockDim.x + threadIdx.x;
    if (t >= total) return;
    int l = t % LIN_IN;
    int b = t / LIN_IN;
    float v = bias[0];
    const float* xb = x + (size_t)b * CC * LIN_IN;
    for (int c = 0; c < CC; c++) {
        const float* xc = xb + c * LIN_IN;
        const float* wc = w + c * 3;
        #pragma unroll
        for (int tt = 0; tt < 3; tt++) {
            int li = l + tt - 1;
            int liC = min(max(li, 0), LIN_IN - 1);
            float msk = (li >= 0 && li < LIN_IN) ? 1.f : 0.f;
            v += wc[tt] * xc[liC] * msk;
        }
    }
    y[t] = v;
}

// head: linear (B,248) @ (400,248)^T + b
__global__ void k_head2(const float* __restrict__ y, const float* __restrict__ lw,
                        const float* __restrict__ lb, float* __restrict__ out, int total)
{
    int t = blockIdx.x * blockDim.x + threadIdx.x;
    if (t >= total) return;
    int o = t % OUTL;
    int b = t / OUTL;
    float v = lb[o];
    const float* yb = y + b * LIN_IN;
    const float* w = lw + (size_t)o * LIN_IN;
    #pragma unroll 8
    for (int l = 0; l < LIN_IN; l++) v += yb[l] * w[l];
    out[t] = v;
}

// =================================================================================
extern "C" void kernel_launch(void* const* d_in, const int* in_sizes, int n_in,
                              void* d_out, int out_size, void* d_ws, size_t ws_size,
                              hipStream_t stream)
{
    (void)in_sizes; (void)n_in; (void)out_size; (void)ws_size;
    const float* snippet  = (const float*)d_in[0];
    const int*   cid      = (const int*)  d_in[1];
    const float* conv0a_w = (const float*)d_in[2];
    const float* conv0a_b = (const float*)d_in[3];
    const float* conv0b_w = (const float*)d_in[4];
    const float* bn0_g    = (const float*)d_in[5];
    const float* bn0_b    = (const float*)d_in[6];
    const float* vae_mu   = (const float*)d_in[7];
    const float* vae_lv   = (const float*)d_in[8];
    const float* qd_w1    = (const float*)d_in[9];
    const float* qd_b1    = (const float*)d_in[10];
    const float* qd_w2    = (const float*)d_in[11];
    const float* qd_b2    = (const float*)d_in[12];
    const float* qd_w3    = (const float*)d_in[13];
    const float* qd_b3    = (const float*)d_in[14];
    const float* wh_keys  = (const float*)d_in[15];
    const float* wh_w     = (const float*)d_in[16];
    const float* wh_b     = (const float*)d_in[17];
    const float* b0_w1    = (const float*)d_in[18];
    const float* rest_w1  = (const float*)d_in[19];
    const float* bn1_g    = (const float*)d_in[20];
    const float* bn1_b    = (const float*)d_in[21];
    const float* bn2_g    = (const float*)d_in[22];
    const float* bn2_b    = (const float*)d_in[23];
    const float* w2       = (const float*)d_in[24];
    const float* bn3_g    = (const float*)d_in[25];
    const float* bn3_b    = (const float*)d_in[26];
    const float* sc0_w    = (const float*)d_in[27];
    const float* sc23_w   = (const float*)d_in[28];
    const float* sc_g     = (const float*)d_in[29];
    const float* sc_b     = (const float*)d_in[30];
    const float* l4_w     = (const float*)d_in[31];
    const float* l4_b     = (const float*)d_in[32];
    const float* lin_w    = (const float*)d_in[33];
    const float* lin_b    = (const float*)d_in[34];

    float* out = (float*)d_out;               // [ (B,400) | z_mu(B,2) | z_logvar(B,2) ]

    // ---- workspace carve ----
    float* ws = (float*)d_ws;
    size_t off = 0;
    auto carve = [&](size_t n) { float* p = ws + off; off += n; return p; };
    float* x0a  = carve((size_t)BB * C0 * L0);
    float* x0   = carve((size_t)BB * C0 * L0);
    float* h2   = carve((size_t)BB * 30);
    float* qrs  = carve((size_t)BB * NQ * KEYL);
    float* dW   = carve((size_t)BB * NQ * KD);
    float* db   = carve((size_t)BB * NQ);
    float* hY   = carve((size_t)BB * LIN_IN);
    float* bufA = carve((size_t)BB * CC * L0);
    float* bufB = carve((size_t)BB * CC * L0);
    float* bufC = carve((size_t)BB * CC * L0);

    // ---- layer0 ----
    {
        int tot = BB * C0 * L0;
        k_conv0a<<<(tot + 255) / 256, 256, 0, stream>>>(snippet, conv0a_w, conv0a_b, x0a, tot);
        dim3 gb(BB * (L0 / 16), 2);
        k_conv0b_wmma<<<gb, 32, 0, stream>>>(x0a, conv0b_w, bn0_g, bn0_b, x0);
    }
    // ---- VAE outputs + query decoder + warehouse ----
    k_z<<<(BB * ZD + 255) / 256, 256, 0, stream>>>(cid, vae_mu, vae_lv,
                                                   out + BB * OUTL, out + BB * OUTL + BB * ZD);
    k_qd<<<BB, 32, 0, stream>>>(cid, vae_mu, qd_w1, qd_b1, qd_w2, qd_b2, h2);
    {
        int tot = BB * NQ * KEYL;
        k_queries<<<(tot + 255) / 256, 256, 0, stream>>>(h2, qd_w3, qd_b3, qrs, tot);
    }
    k_warehouse<<<(BB * NQ) / (QPW * 8), 256, 0, stream>>>(qrs, wh_keys, wh_w, wh_b, dW, db);

    // ---- ResBlock stack ----
    const int Lins[NB]    = {992, 992, 992, 496, 248, 248, 248};
    const int strides[NB] = {1, 1, 2, 2, 1, 1, 1};
    for (int i = 0; i < NB; i++) {
        int Lin = Lins[i], s = strides[i];
        int Lout = (s == 2) ? Lin / 2 : Lin;
        const float* Xin = (i == 0) ? x0 : bufA;
        const float* W1 = (i == 0) ? b0_w1 : rest_w1 + (size_t)(i - 1) * CC * CC;

        // gemm1: y1 = lrelu(bn1(W1 @ x))
        int ntIn = (Lin + 15) / 16;
        dim3 g1(BB * ntIn, (CC + 15) / 16);
        if (i == 0)
            k_gemm<1, 0, C0><<<g1, 32, 0, stream>>>(W1, CC, Xin, Lin, 1,
                                                    bn1_g + i * CC, bn1_b + i * CC,
                                                    nullptr, bufB, Lin, ntIn);
        else
            k_gemm<1, 0, CC><<<g1, 32, 0, stream>>>(W1, CC, Xin, Lin, 1,
                                                    bn1_g + i * CC, bn1_b + i * CC,
                                                    nullptr, bufB, Lin, ntIn);
        // dynamic depthwise + bn2 + lrelu
        int totdw = BB * CC * Lout;
        k_dw<<<(totdw + 255) / 256, 256, 0, stream>>>(bufB, dW, db, i * CC,
                                                      bn2_g + i * CC, bn2_b + i * CC,
                                                      bufC, Lin, Lout, s, totdw);
        // shortcut
        const float* shortPtr;
        int ntOut = (Lout + 15) / 16;
        if (i == 0 || i == 2 || i == 3) {
            int j = (i == 0) ? 0 : (i == 2 ? 1 : 2);
            const float* scw = (i == 0) ? sc0_w : sc23_w + (size_t)(j - 1) * CC * CC;
            dim3 gs(BB * ntOut, (CC + 15) / 16);
            if (i == 0)
                k_gemm<0, 0, C0><<<gs, 32, 0, stream>>>(scw, CC, Xin, Lin, s,
                                                        sc_g + j * CC, sc_b + j * CC,
                                                        nullptr, bufB, Lout, ntOut);
            else
                k_gemm<0, 0, CC><<<gs, 32, 0, stream>>>(scw, CC, Xin, Lin, s,
                                                        sc_g + j * CC, sc_b + j * CC,
                                                        nullptr, bufB, Lout, ntOut);
            shortPtr = bufB;
        } else {
            shortPtr = bufA;   // identity (aliases Y; per-element read-before-write is safe)
        }
        // gemm2: x = lrelu(bn3(W2 @ hh) + short)
        dim3 g2(BB * ntOut, (CC + 15) / 16);
        k_gemm<1, 1, CC><<<g2, 32, 0, stream>>>(w2 + (size_t)i * CC * CC, CC, bufC, Lout, 1,
                                                bn3_g + i * CC, bn3_b + i * CC,
                                                shortPtr, bufA, Lout, ntOut);
    }

    // ---- head ----
    {
        int tot1 = BB * LIN_IN;
        k_head1<<<(tot1 + 255) / 256, 256, 0, stream>>>(bufA, l4_w, l4_b, hY, tot1);
        int tot2 = BB * OUTL;
        k_head2<<<(tot2 + 255) / 256, 256, 0, stream>>>(hY, lin_w, lin_b, out, tot2);
    }
}